// MlpMoeBlock_30056181137726
// MI455X (gfx1250) — compile-verified
//
#include <hip/hip_runtime.h>
#include <cstddef>
#include <cstdint>

typedef float v2f __attribute__((ext_vector_type(2)));
typedef float v4f __attribute__((ext_vector_type(4)));
typedef float v8f __attribute__((ext_vector_type(8)));
typedef int   v4i __attribute__((ext_vector_type(4)));

constexpr int Bb    = 8;
constexpr int S     = 1024;
constexpr int D     = 1024;
constexpr int E     = 8;
constexpr int M     = 4096;
constexpr int GROUP = 1024;
constexpr int G     = (Bb * S) / GROUP;   // 8
constexpr int CAP   = 256;                // ceil(GROUP*K/E)
constexpr int NTOK  = G * GROUP;          // 8192
constexpr int NROWS = E * G * CAP;        // 16384

// ---- CDNA5 fast tanh (V_TANH_F32) if the toolchain exposes it -------------
#if __has_builtin(__builtin_amdgcn_tanhf)
#define FAST_TANH(x) __builtin_amdgcn_tanhf(x)
#elif __has_builtin(__builtin_amdgcn_tanh_f32)
#define FAST_TANH(x) __builtin_amdgcn_tanh_f32(x)
#else
#define FAST_TANH(x) tanhf(x)
#endif

// ---- CDNA5 async global->LDS copy (GLOBAL_LOAD_ASYNC_TO_LDS_B128) ---------
// Probe-confirmed signature: (as1 v4i* gsrc, as3 v4i* ldst, imm offset, imm cpol)
#if __has_builtin(__builtin_amdgcn_global_load_async_to_lds_b128)
#define HAVE_ASYNC_LDS 1
typedef __attribute__((address_space(1))) v4i g_v4i;
typedef __attribute__((address_space(3))) v4i l_v4i;
#endif

static __device__ __forceinline__ void wait_async_all()
{
#if __has_builtin(__builtin_amdgcn_s_wait_asynccnt)
    __builtin_amdgcn_s_wait_asynccnt(0);
#else
    asm volatile("s_wait_asynccnt 0" ::: "memory");
#endif
}

// ---------------------------------------------------------------------------
// K0: router — logits, softmax, top-2, deterministic imp partial sums
// grid: NTOK/256 blocks of 256 threads; one thread per token
// ---------------------------------------------------------------------------
__global__ __launch_bounds__(256) void router_kernel(
    const float* __restrict__ x, const float* __restrict__ wr,
    int* __restrict__ tok_top, float* __restrict__ tok_gate,
    float* __restrict__ partial_imp)
{
    __shared__ float red[256][8];
    const int tid = threadIdx.x;
    const int t   = blockIdx.x * 256 + tid;

    float acc[8];
#pragma unroll
    for (int e = 0; e < 8; ++e) acc[e] = 0.f;

    const float4* xr = (const float4*)(x + (size_t)t * D);
    for (int d4 = 0; d4 < D / 4; ++d4) {
        float4 xv = xr[d4];
        float xs[4] = {xv.x, xv.y, xv.z, xv.w};
#pragma unroll
        for (int j = 0; j < 4; ++j) {
            const float* w = wr + (size_t)(d4 * 4 + j) * E;
#pragma unroll
            for (int e = 0; e < 8; ++e) acc[e] += xs[j] * w[e];
        }
    }
    // softmax over E=8
    float m = acc[0];
#pragma unroll
    for (int e = 1; e < 8; ++e) m = fmaxf(m, acc[e]);
    float p[8], s = 0.f;
#pragma unroll
    for (int e = 0; e < 8; ++e) { p[e] = __expf(acc[e] - m); s += p[e]; }
    const float inv = 1.f / s;
#pragma unroll
    for (int e = 0; e < 8; ++e) p[e] *= inv;

    // top-2 (first occurrence wins ties, matching lax.top_k)
    int i0 = 0; float v0 = p[0];
#pragma unroll
    for (int e = 1; e < 8; ++e) if (p[e] > v0) { v0 = p[e]; i0 = e; }
    int i1 = (i0 == 0) ? 1 : 0; float v1 = p[i1];
#pragma unroll
    for (int e = 0; e < 8; ++e)
        if (e != i0 && p[e] > v1) { v1 = p[e]; i1 = e; }

    tok_top[t * 2 + 0]  = i0;  tok_top[t * 2 + 1]  = i1;
    tok_gate[t * 2 + 0] = v0;  tok_gate[t * 2 + 1] = v1;

    // deterministic imp reduction (fixed-order tree)
#pragma unroll
    for (int e = 0; e < 8; ++e) red[tid][e] = p[e];
    __syncthreads();
    for (int off = 128; off > 0; off >>= 1) {
        if (tid < off)
#pragma unroll
            for (int e = 0; e < 8; ++e) red[tid][e] += red[tid + off][e];
        __syncthreads();
    }
    if (tid == 0)
#pragma unroll
        for (int e = 0; e < 8; ++e) partial_imp[blockIdx.x * 8 + e] = red[0][e];
}

// ---------------------------------------------------------------------------
// K1: capacity assignment. One block per group; 8 threads replay the
// reference cumsum semantics (positions count ALL tokens, k=1 offset by
// k=0's full per-expert count).
// ---------------------------------------------------------------------------
__global__ __launch_bounds__(256) void assign_kernel(
    const int* __restrict__ tok_top, const float* __restrict__ tok_gate,
    int* __restrict__ token_slot, int* __restrict__ tok_dst,
    float* __restrict__ tok_w)
{
    __shared__ int   stop[2][GROUP];
    __shared__ float sg[2][GROUP];
    const int g = blockIdx.x, tid = threadIdx.x;

    for (int i = tid; i < GROUP; i += 256) {
        int t = g * GROUP + i;
        stop[0][i] = tok_top[t * 2];     stop[1][i] = tok_top[t * 2 + 1];
        sg[0][i]   = tok_gate[t * 2];    sg[1][i]   = tok_gate[t * 2 + 1];
    }
    for (int i = tid; i < E * CAP; i += 256) {
        int e = i / CAP, c = i % CAP;
        token_slot[(e * G + g) * CAP + c] = -1;
    }
    __syncthreads();

    if (tid < E) {
        const int e = tid;
        int cnt0 = 0;
        for (int s = 0; s < GROUP; ++s) {
            if (stop[0][s] == e) {
                int pos = cnt0++;
                int t = g * GROUP + s;
                if (pos < CAP) {
                    token_slot[(e * G + g) * CAP + pos] = s;
                    tok_dst[t * 2] = (e << 16) | pos;  tok_w[t * 2] = sg[0][s];
                } else { tok_dst[t * 2] = -1; tok_w[t * 2] = 0.f; }
            }
        }
        int cnt1 = 0;
        for (int s = 0; s < GROUP; ++s) {
            if (stop[1][s] == e) {
                int pos = cnt0 + cnt1++;
                int t = g * GROUP + s;
                if (pos < CAP) {
                    token_slot[(e * G + g) * CAP + pos] = s;
                    tok_dst[t * 2 + 1] = (e << 16) | pos;  tok_w[t * 2 + 1] = sg[1][s];
                } else { tok_dst[t * 2 + 1] = -1; tok_w[t * 2 + 1] = 0.f; }
            }
        }
    }
}

// ---------------------------------------------------------------------------
// K2: fused expert MLP, fp32 WMMA (v_wmma_f32_16x16x4_f32).
// 1 workgroup (8 waves) = 16 expert slots. h never leaves LDS.
// ---------------------------------------------------------------------------
__global__ __launch_bounds__(256) void moe_mlp_kernel(
    const float* __restrict__ x,  const float* __restrict__ w1,
    const float* __restrict__ b1, const float* __restrict__ w2,
    const float* __restrict__ b2, const int* __restrict__ token_slot,
    float* __restrict__ ye)
{
    constexpr int AS = 1028;   // padded A stride (bank-conflict-free: 1028%64==4)
    constexpr int HS = 132;    // padded h stride
    constexpr int MC = 128;    // M-chunk
    __shared__ float A_s[16 * AS];   // 16 x 1024 gathered x rows (~64KB)
    __shared__ float h_s[16 * HS];   // 16 x 128 gelu'd hidden chunk (~8KB)

    const int tid = threadIdx.x;
    const int r0  = blockIdx.x * 16;        // first of 16 slot-rows
    const int e   = r0 >> 11;               // G*CAP = 2048 rows/expert
    const int g   = (r0 >> 8) & 7;          // CAP = 256 rows/group

    // gather 16 rows of x (or zeros for empty slots) into LDS
#ifdef HAVE_ASYNC_LDS
    for (int row = 0; row < 16; ++row) {
        int s = token_slot[r0 + row];       // block-uniform -> EXEC stays full
        float* ldsp = &A_s[row * AS + tid * 4];
        if (s >= 0) {
            const float* gp = x + (size_t)(g * GROUP + s) * D + tid * 4;
            __builtin_amdgcn_global_load_async_to_lds_b128(
                (g_v4i*)gp, (l_v4i*)ldsp, 0, 0);
        } else {
            float4 z = {0.f, 0.f, 0.f, 0.f};
            *(float4*)ldsp = z;
        }
    }
    wait_async_all();
#else
    for (int row = 0; row < 16; ++row) {
        int s = token_slot[r0 + row];
        float4 v = {0.f, 0.f, 0.f, 0.f};
        if (s >= 0)
            v = ((const float4*)(x + (size_t)(g * GROUP + s) * D))[tid];
        *(float4*)&A_s[row * AS + tid * 4] = v;
    }
#endif
    __syncthreads();

    const int lane = tid & 31, w = tid >> 5;
    const int hi = lane >> 4, n = lane & 15;   // A frag: row=n, K-half=hi
    const float* w1e = w1 + (size_t)e * D * M;
    const float* w2e = w2 + (size_t)e * M * D;

    v8f zero = {};
    v8f acc2[8];                                // 16 x 128 ye slice per wave
#pragma unroll
    for (int i = 0; i < 8; ++i) acc2[i] = zero;

    for (int mc = 0; mc < M / MC; ++mc) {
        // ---- GEMM1: this wave's 16-col h subtile over K=D=1024 ----
        const int mcol = mc * MC + w * 16 + n;
        // prefetch the next chunk's first w1 line (global_prefetch_b8)
        if (mc + 1 < M / MC)
            __builtin_prefetch(w1e + (size_t)(2 * hi) * M + mcol + MC, 0, 1);
        v8f c1 = zero;
#pragma unroll 4
        for (int kk = 0; kk < D / 4; ++kk) {
            const int k = kk * 4 + 2 * hi;
            v2f a, b;
            a.x = A_s[n * AS + k];  a.y = A_s[n * AS + k + 1];
            const float* bp = w1e + (size_t)k * M + mcol;
            b.x = bp[0];            b.y = bp[M];
            c1 = __builtin_amdgcn_wmma_f32_16x16x4_f32(
                     false, a, false, b, (short)0, c1, false, false);
        }
        const float bias1 = b1[e * M + mcol];
        __syncthreads();           // WAR: prior gemm2 reads of h_s done
#pragma unroll
        for (int i = 0; i < 8; ++i) {
            float v = c1[i] + bias1;                      // tanh-gelu
            float u = 0.7978845608028654f * (v + 0.044715f * v * v * v);
            h_s[(i + 8 * hi) * HS + w * 16 + n] = 0.5f * v * (1.0f + FAST_TANH(u));
        }
        __syncthreads();           // h chunk ready

        // ---- GEMM2: accumulate ye[16 x (w*128..w*128+127)] over K=MC ----
#pragma unroll 1
        for (int t8 = 0; t8 < 8; ++t8) {
            const int dcol = w * 128 + t8 * 16 + n;
#pragma unroll 4
            for (int kk = 0; kk < MC / 4; ++kk) {
                const int k = kk * 4 + 2 * hi;
                v2f a, b;
                a.x = h_s[n * HS + k];  a.y = h_s[n * HS + k + 1];
                const float* bp = w2e + (size_t)(mc * MC + k) * D + dcol;
                b.x = bp[0];            b.y = bp[D];
                acc2[t8] = __builtin_amdgcn_wmma_f32_16x16x4_f32(
                               false, a, false, b, (short)0, acc2[t8], false, false);
            }
        }
    }

    // store ye tile (+b2). C/D layout: VGPR i -> rows i / i+8 by lane-half.
    for (int t8 = 0; t8 < 8; ++t8) {
        const int dcol = w * 128 + t8 * 16 + n;
        const float bb = b2[e * D + dcol];
#pragma unroll
        for (int i = 0; i < 8; ++i)
            ye[(size_t)(r0 + i + 8 * hi) * D + dcol] = acc2[t8][i] + bb;
    }
}

// ---------------------------------------------------------------------------
// K3: combine — out[t] = sum_k w_k * ye[row_k]; fixed add order (determinism)
// ---------------------------------------------------------------------------
__global__ __launch_bounds__(256) void combine_kernel(
    const float* __restrict__ ye, const int* __restrict__ tok_dst,
    const float* __restrict__ tok_w, float* __restrict__ out)
{
    const int t = blockIdx.x, tid = threadIdx.x;
    const int g = t >> 10;
    v4f acc = {0.f, 0.f, 0.f, 0.f};
#pragma unroll
    for (int k = 0; k < 2; ++k) {
        int dst = tok_dst[t * 2 + k];
        if (dst >= 0) {
            int ee = dst >> 16, c = dst & 0xffff;
            float wv = tok_w[t * 2 + k];
            v4f yv = ((const v4f*)(ye + (size_t)((ee * G + g) * CAP + c) * D))[tid];
            acc += wv * yv;
        }
    }
    // output is written once and never re-read by the GPU: non-temporal store
    __builtin_nontemporal_store(acc, (v4f*)(out + (size_t)t * D) + tid);
}

// ---------------------------------------------------------------------------
// K4: aux = mean_g (std(imp)/mean(imp))^2, appended after the main output
// ---------------------------------------------------------------------------
__global__ void aux_kernel(const float* __restrict__ partial_imp,
                           float* __restrict__ out)
{
    if (threadIdx.x == 0 && blockIdx.x == 0) {
        float aux = 0.f;
        for (int g = 0; g < G; ++g) {
            float imp[8], mean = 0.f;
            for (int e = 0; e < 8; ++e) {
                float v = 0.f;
                for (int b = 0; b < 4; ++b) v += partial_imp[(g * 4 + b) * 8 + e];
                imp[e] = v; mean += v;
            }
            mean *= 0.125f;
            float var = 0.f;
            for (int e = 0; e < 8; ++e) { float d = imp[e] - mean; var += d * d; }
            var *= 0.125f;                        // population std (jnp default)
            aux += var / (mean * mean);
        }
        out[(size_t)NTOK * D] = aux * 0.125f;     // mean over G=8
    }
}

// ---------------------------------------------------------------------------
extern "C" void kernel_launch(void* const* d_in, const int* in_sizes, int n_in,
                              void* d_out, int out_size, void* d_ws, size_t ws_size,
                              hipStream_t stream)
{
    const float* x  = (const float*)d_in[0];
    const float* wr = (const float*)d_in[1];
    const float* w1 = (const float*)d_in[2];
    const float* b1 = (const float*)d_in[3];
    const float* w2 = (const float*)d_in[4];
    const float* b2 = (const float*)d_in[5];
    float* out = (float*)d_out;

    // workspace layout (~64.3 MB)
    float* ye          = (float*)d_ws;                       // NROWS*D
    int*   token_slot  = (int*)(ye + (size_t)NROWS * D);     // NROWS
    int*   tok_top     = token_slot + NROWS;                 // NTOK*2
    float* tok_gate    = (float*)(tok_top + NTOK * 2);       // NTOK*2
    int*   tok_dst     = (int*)(tok_gate + NTOK * 2);        // NTOK*2
    float* tok_w       = (float*)(tok_dst + NTOK * 2);       // NTOK*2
    float* partial_imp = tok_w + NTOK * 2;                   // (NTOK/256)*8

    router_kernel <<<NTOK / 256, 256, 0, stream>>>(x, wr, tok_top, tok_gate, partial_imp);
    assign_kernel <<<G,          256, 0, stream>>>(tok_top, tok_gate, token_slot, tok_dst, tok_w);
    moe_mlp_kernel<<<NROWS / 16, 256, 0, stream>>>(x, w1, b1, w2, b2, token_slot, ye);
    combine_kernel<<<NTOK,       256, 0, stream>>>(ye, tok_dst, tok_w, out);
    aux_kernel    <<<1,           32, 0, stream>>>(partial_imp, out);
}